// LatticeSnake_37933151158341
// MI455X (gfx1250) — compile-verified
//
#include <hip/hip_runtime.h>

typedef float v2f __attribute__((ext_vector_type(2)));
typedef float v8f __attribute__((ext_vector_type(8)));

namespace {
constexpr int NRES  = 64;          // N
constexpr int NPTS  = 2 * NRES - 1;  // 127 points per sample
constexpr int NSLOT = 128;           // padded to 128 (32 K-steps of 4)
constexpr int CELLS = 343;           // 7*7*7 window cells
constexpr int LPAD  = 323;           // LP sentinel for masked points
}

// One wave (32 lanes) per (b, n) window.
// out[c] = sum_p ind[c,p] * v[p] computed as a 16xK WMMA indicator matvec,
// kept in exact f32 via V_WMMA_F32_16X16X4_F32.
__global__ __launch_bounds__(32)
void lattice_snake_wmma(const float* __restrict__ acids,        // B x 64
                        const unsigned char* __restrict__ mask, // B x 64 (bool)
                        const int* __restrict__ idx,            // B x 64 x 3
                        float* __restrict__ out)                // B x 64 x 343
{
  const int bn   = blockIdx.x;           // 0 .. B*N-1
  const int b    = bn >> 6;
  const int n    = bn & (NRES - 1);
  const int lane = threadIdx.x;          // 0 .. 31

  __shared__ int   s_cell[NSLOT];
  __shared__ float s_val[NSLOT];

  const int*           idxb = idx   + b * NRES * 3;
  const float*         acb  = acids + b * NRES;
  const unsigned char* mkb  = mask  + b * NRES;

  // Window start = idx2[b][n][:] = 2*(idx+63)
  const int st0 = 2 * (idxb[n * 3 + 0] + (NRES - 1));
  const int st1 = 2 * (idxb[n * 3 + 1] + (NRES - 1));
  const int st2 = 2 * (idxb[n * 3 + 2] + (NRES - 1));

  // Build the 128-slot point table (cell id within this window, value).
  #pragma unroll
  for (int t = 0; t < 4; ++t) {
    const int p = lane * 4 + t;          // 0..127
    int   cell = -1;
    float val  = 0.f;
    int a0 = LPAD, a1 = LPAD, a2 = LPAD;
    bool live = false;
    if (p < NRES) {
      // direct point: ai = 2*(idx+63)+HALF = 2*idx + 129
      a0 = 2 * idxb[p * 3 + 0] + 129;
      a1 = 2 * idxb[p * 3 + 1] + 129;
      a2 = 2 * idxb[p * 3 + 2] + 129;
      if (!mkb[p]) { a0 = a1 = a2 = LPAD; }
      val  = acb[p];
      live = true;
    } else if (p < NPTS) {
      // midpoint q,q+1: ai = (idx2_q + idx2_{q+1})/2 + HALF = idx_q + idx_{q+1} + 129
      const int q = p - NRES;
      a0 = idxb[q * 3 + 0] + idxb[(q + 1) * 3 + 0] + 129;
      a1 = idxb[q * 3 + 1] + idxb[(q + 1) * 3 + 1] + 129;
      a2 = idxb[q * 3 + 2] + idxb[(q + 1) * 3 + 2] + 129;
      if (!mkb[q + 1]) { a0 = a1 = a2 = LPAD; }
      val  = acb[q] + acb[q + 1] + 1.0f;
      live = true;
    }
    if (live) {
      const int o0 = a0 - st0, o1 = a1 - st1, o2 = a2 - st2;
      if ((unsigned)o0 < 7u && (unsigned)o1 < 7u && (unsigned)o2 < 7u)
        cell = (o0 * 7 + o1) * 7 + o2;
    }
    s_cell[p] = cell;
    s_val[p]  = val;
  }
  __syncthreads();

  float* outbn = out + bn * CELLS;

  // Masked window -> all zeros (block-uniform branch; no WMMA executes).
  if (!mkb[n]) {
    for (int c = lane; c < CELLS; c += 32) outbn[c] = 0.f;
    return;
  }

  const int h = lane >> 4;     // lane half: K offset 2h (A and B share this)
  const int r = lane & 15;     // A row (M) / B column (N)

  #pragma unroll 1
  for (int mt = 0; mt < 22; ++mt) {        // 22 tiles of 16 cells cover 343
    v8f acc = {0.f, 0.f, 0.f, 0.f, 0.f, 0.f, 0.f, 0.f};
    const int myrow = mt * 16 + r;
    #pragma unroll
    for (int kk = 0; kk < 32; ++kk) {      // K = 4 per step, 128 slots total
      const int base = kk * 4 + 2 * h;
      const int c0 = s_cell[base];
      const int c1 = s_cell[base + 1];
      v2f A, Bv;
      A.x  = (c0 == myrow) ? 1.f : 0.f;    // A[M=r, K=2h+0]
      A.y  = (c1 == myrow) ? 1.f : 0.f;    // A[M=r, K=2h+1]
      Bv.x = s_val[base];                  // B[K=2h+0, all 16 cols]
      Bv.y = s_val[base + 1];              // B[K=2h+1, all 16 cols]
      acc = __builtin_amdgcn_wmma_f32_16x16x4_f32(
          /*neg_a=*/false, A, /*neg_b=*/false, Bv,
          /*c_mod=*/(short)0, acc, /*reuse_a=*/false, /*reuse_b=*/false);
    }
    // D layout: lane half h, VGPR vg -> M = vg + 8h (all columns identical).
    if (r == 0) {
      const int cbase = mt * 16 + 8 * h;
      #pragma unroll
      for (int vg = 0; vg < 8; ++vg) {
        const int c = cbase + vg;
        if (c < CELLS) outbn[c] = acc[vg];
      }
    }
  }
}

extern "C" void kernel_launch(void* const* d_in, const int* in_sizes, int n_in,
                              void* d_out, int out_size, void* d_ws, size_t ws_size,
                              hipStream_t stream) {
  const float*         acids = (const float*)d_in[0];
  const unsigned char* mask  = (const unsigned char*)d_in[1];
  const int*           idx   = (const int*)d_in[2];
  float*               out   = (float*)d_out;

  const int BN = in_sizes[0];  // B*N = 128 windows, one wave each
  lattice_snake_wmma<<<BN, 32, 0, stream>>>(acids, mask, idx, out);
}